// EuclideanGATLayer_27273042329633
// MI455X (gfx1250) — compile-verified
//
#include <hip/hip_runtime.h>
#include <math.h>

#define IN_DIM   128
#define OUT_DIM  128
#define NEG_SLOPE 0.2f

typedef float v2f __attribute__((ext_vector_type(2)));
typedef float v8f __attribute__((ext_vector_type(8)));

// ---------------------------------------------------------------------------
// Kernel 1: h = x @ W with fp32 WMMA (V_WMMA_F32_16X16X4_F32).
// One wave computes a 16-row slab of h across all 8 column tiles.
// A layout (16x4 f32): lane 0-15 -> M=lane, K=k0+{0,1}; lane 16-31 -> K=k0+{2,3}
// B layout (4x16 f32): VGPR0 = rows {k0, k0+2}, VGPR1 = rows {k0+1, k0+3}, col = lane&15
// C/D layout: VGPR r -> M = r + 8*(lane>=16), N = lane&15
// ---------------------------------------------------------------------------
__global__ __launch_bounds__(256) void gat_gemm_wmma(const float* __restrict__ x,
                                                     const float* __restrict__ W,
                                                     float* __restrict__ h, int N) {
    const int wave = (int)((blockIdx.x * blockDim.x + threadIdx.x) >> 5);
    const int lane = threadIdx.x & 31;
    const int row0 = wave * 16;
    if (row0 >= N) return;                       // wave-uniform: EXEC stays all-ones

    const int mrow = row0 + (lane & 15);
    const int koff = (lane >> 4) << 1;           // 0 or 2
    const int bl   = lane & 15;

    v8f acc[8];
#pragma unroll
    for (int t = 0; t < 8; ++t) acc[t] = (v8f){0, 0, 0, 0, 0, 0, 0, 0};

    const float* __restrict__ arow = x + (size_t)mrow * IN_DIM;
#pragma unroll 4
    for (int k0 = 0; k0 < IN_DIM; k0 += 4) {
        v2f a;
        a.x = arow[k0 + koff + 0];
        a.y = arow[k0 + koff + 1];
        const float* __restrict__ w0 = W + (size_t)(k0 + koff) * OUT_DIM;
#pragma unroll
        for (int t = 0; t < 8; ++t) {
            v2f b;
            b.x = w0[t * 16 + bl];
            b.y = w0[OUT_DIM + t * 16 + bl];
            acc[t] = __builtin_amdgcn_wmma_f32_16x16x4_f32(
                false, a, false, b, (short)0, acc[t], false, false);
        }
    }

    const int orow = row0 + ((lane >> 4) << 3);
#pragma unroll
    for (int t = 0; t < 8; ++t) {
        float* hp = h + (size_t)orow * OUT_DIM + t * 16 + bl;
#pragma unroll
        for (int r = 0; r < 8; ++r) hp[(size_t)r * OUT_DIM] = acc[t][r];
    }
}

// ---------------------------------------------------------------------------
// Kernel 2: per-node attention logits e_src = h.a_src, e_dst = h.a_dst
// One wave per node, 4 coalesced elements per lane, shuffle reduction.
// ---------------------------------------------------------------------------
__global__ __launch_bounds__(256) void gat_scores(const float* __restrict__ h,
                                                  const float* __restrict__ a_src,
                                                  const float* __restrict__ a_dst,
                                                  float* __restrict__ e_src,
                                                  float* __restrict__ e_dst, int N) {
    const int node = (int)((blockIdx.x * blockDim.x + threadIdx.x) >> 5);
    const int lane = threadIdx.x & 31;
    if (node >= N) return;
    const float* __restrict__ hr = h + (size_t)node * OUT_DIM;
    float s = 0.f, d = 0.f;
#pragma unroll
    for (int j = 0; j < OUT_DIM; j += 32) {
        float v = hr[j + lane];
        s += v * a_src[j + lane];
        d += v * a_dst[j + lane];
    }
#pragma unroll
    for (int off = 16; off > 0; off >>= 1) {
        s += __shfl_down(s, off, 32);
        d += __shfl_down(d, off, 32);
    }
    if (lane == 0) { e_src[node] = s; e_dst[node] = d; }
}

// ---------------------------------------------------------------------------
// Kernel 0: init accumulators (agg=0 in d_out, recv_max=-inf, denom=0)
// ---------------------------------------------------------------------------
__global__ void gat_init(float* __restrict__ agg, float* __restrict__ recv_max,
                         float* __restrict__ denom, int N) {
    int i = blockIdx.x * blockDim.x + threadIdx.x;
    int total = N * OUT_DIM;
    if (i < total) agg[i] = 0.f;
    if (i < N) { recv_max[i] = -INFINITY; denom[i] = 0.f; }
}

// float atomic max via int punning (nonneg -> int max, neg -> uint min)
__device__ __forceinline__ void atomicMaxFloat(float* addr, float val) {
    if (val >= 0.f) atomicMax((int*)addr, __float_as_int(val));
    else            atomicMin((unsigned int*)addr, __float_as_uint(val));
}

// ---------------------------------------------------------------------------
// Kernel 3: raw edge scores + segment max over destinations
// ---------------------------------------------------------------------------
__global__ void gat_edge_raw(const long long* __restrict__ src,
                             const long long* __restrict__ dst,
                             const float* __restrict__ e_src,
                             const float* __restrict__ e_dst,
                             float* __restrict__ raw,
                             float* __restrict__ recv_max, int E) {
    int e = blockIdx.x * blockDim.x + threadIdx.x;
    if (e >= E) return;
    int s = (int)src[e], d = (int)dst[e];
    float r = e_src[s] + e_dst[d];
    r = (r > 0.f) ? r : NEG_SLOPE * r;
    raw[e] = r;
    atomicMaxFloat(&recv_max[d], r);
}

// ---------------------------------------------------------------------------
// Kernel 4: exp(raw - max[dst]) + segment sum (denominator)
// ---------------------------------------------------------------------------
__global__ void gat_edge_exp(const long long* __restrict__ dst,
                             const float* __restrict__ recv_max,
                             float* __restrict__ raw_exp,      // in: raw, out: exp
                             float* __restrict__ denom, int E) {
    int e = blockIdx.x * blockDim.x + threadIdx.x;
    if (e >= E) return;
    int d = (int)dst[e];
    float m = recv_max[d];
    if (!isfinite(m)) m = 0.f;
    float ex = expf(raw_exp[e] - m);
    raw_exp[e] = ex;
    atomicAdd(&denom[d], ex);
}

// ---------------------------------------------------------------------------
// Kernel 5: weighted scatter  agg[dst] += alpha * h[src]
// 32 lanes per edge, float4 gather (coalesced 128-float rows), f32 atomics.
// ---------------------------------------------------------------------------
__global__ __launch_bounds__(256) void gat_edge_scatter(const long long* __restrict__ src,
                                                        const long long* __restrict__ dst,
                                                        const float* __restrict__ h,
                                                        const float* __restrict__ expv,
                                                        const float* __restrict__ denom,
                                                        float* __restrict__ agg, int E) {
    long long t = (long long)blockIdx.x * blockDim.x + threadIdx.x;
    int e = (int)(t >> 5);
    if (e >= E) return;
    int q = ((int)t & 31) * 4;
    int s = (int)src[e], d = (int)dst[e];
    float alpha = expv[e] / fmaxf(denom[d], 1e-15f);
    const float4 hv = *(const float4*)(h + (size_t)s * OUT_DIM + q);
    float* out = agg + (size_t)d * OUT_DIM + q;
    atomicAdd(out + 0, hv.x * alpha);
    atomicAdd(out + 1, hv.y * alpha);
    atomicAdd(out + 2, hv.z * alpha);
    atomicAdd(out + 3, hv.w * alpha);
}

// ---------------------------------------------------------------------------
// Kernel 6: out = elu(agg + bias)   (in place in d_out)
// ---------------------------------------------------------------------------
__global__ void gat_finalize(float* __restrict__ agg, const float* __restrict__ bias,
                             int total) {
    int i = blockIdx.x * blockDim.x + threadIdx.x;
    if (i >= total) return;
    float v = agg[i] + bias[i & (OUT_DIM - 1)];
    agg[i] = (v > 0.f) ? v : expm1f(v);
}

// ---------------------------------------------------------------------------
extern "C" void kernel_launch(void* const* d_in, const int* in_sizes, int n_in,
                              void* d_out, int out_size, void* d_ws, size_t ws_size,
                              hipStream_t stream) {
    const float*     x     = (const float*)d_in[0];
    const long long* eidx  = (const long long*)d_in[1];
    const float*     W     = (const float*)d_in[2];
    const float*     a_src = (const float*)d_in[3];
    const float*     a_dst = (const float*)d_in[4];
    const float*     bias  = (const float*)d_in[5];
    float*           out   = (float*)d_out;

    const int N = in_sizes[0] / IN_DIM;     // 50000
    const int E = in_sizes[1] / 2;          // 800000
    const long long* src = eidx;            // row 0
    const long long* dst = eidx + E;        // row 1

    // workspace layout (all 16B aligned: N*OUT_DIM*4 and N*4 are multiples of 16)
    char* ws = (char*)d_ws;
    float* h        = (float*)ws;                         ws += (size_t)N * OUT_DIM * 4;
    float* e_src    = (float*)ws;                         ws += (size_t)N * 4;
    float* e_dst    = (float*)ws;                         ws += (size_t)N * 4;
    float* recv_max = (float*)ws;                         ws += (size_t)N * 4;
    float* denom    = (float*)ws;                         ws += (size_t)N * 4;
    float* raw_exp  = (float*)ws;                         ws += (size_t)E * 4;

    // init accumulators
    {
        int total = N * OUT_DIM;
        gat_init<<<(total + 255) / 256, 256, 0, stream>>>(out, recv_max, denom, N);
    }
    // h = x @ W (WMMA), one wave per 16 rows -> ceil(N/16) waves, 8 waves/block
    {
        int waves  = (N + 15) / 16;
        int blocks = (waves + 7) / 8;
        gat_gemm_wmma<<<blocks, 256, 0, stream>>>(x, W, h, N);
    }
    // per-node logits, one wave per node
    {
        int blocks = (N + 7) / 8;
        gat_scores<<<blocks, 256, 0, stream>>>(h, a_src, a_dst, e_src, e_dst, N);
    }
    // edge raw + segment max
    gat_edge_raw<<<(E + 255) / 256, 256, 0, stream>>>(src, dst, e_src, e_dst,
                                                      raw_exp, recv_max, E);
    // edge exp + segment sum
    gat_edge_exp<<<(E + 255) / 256, 256, 0, stream>>>(dst, recv_max, raw_exp, denom, E);
    // weighted scatter, 32 lanes per edge
    {
        long long threads = (long long)E * 32;
        int blocks = (int)((threads + 255) / 256);
        gat_edge_scatter<<<blocks, 256, 0, stream>>>(src, dst, h, raw_exp, denom, out, E);
    }
    // elu(agg + bias)
    {
        int total = N * OUT_DIM;
        gat_finalize<<<(total + 255) / 256, 256, 0, stream>>>(out, bias, total);
    }
}